// Model_2585570312255
// MI455X (gfx1250) — compile-verified
//
#include <hip/hip_runtime.h>
#include <math.h>

// ---------------------------------------------------------------------------
// 2-layer RGCN (basis decomposition) for MI455X / gfx1250 (wave32, WMMA).
//  - concept layer: atomic mean-aggregation (bandwidth/L2-atomic bound)
//  - proj GEMMs: f32 WMMA (V_WMMA_F32_16X16X4_F32), root folded into B,
//    K-tiles double-buffered through LDS with GLOBAL_LOAD_ASYNC_TO_LDS
//    (ASYNCcnt-tracked, no VGPR round trip)
//  - edge messages: comp[et] . proj[src] gather + f32 atomics
//  - softmax via wave32 shuffle reductions
// Workspace requirement: ~334 MB (aliased regions).
// ---------------------------------------------------------------------------

typedef __attribute__((ext_vector_type(2))) float v2f;
typedef __attribute__((ext_vector_type(8))) float v8f;
typedef __attribute__((ext_vector_type(4))) int   v4i;

#define AS1 __attribute__((address_space(1)))
#define AS3 __attribute__((address_space(3)))

#define N_G1 120000
#define N_G2 100000

// ----------------------------- degree count --------------------------------
__global__ void k_degree(const int* __restrict__ ei, int E,
                         float* __restrict__ deg, int nmax) {
  int e = blockIdx.x * blockDim.x + threadIdx.x;
  if (e >= E) return;
  int dst = ei[E + e];
  if (dst < nmax) atomicAdd(&deg[dst], 1.0f);
}

// ------------------------- concept layer aggregate -------------------------
// one wave per edge; each lane moves 4 consecutive floats of the 128-f row
__global__ void k_concept_agg(const int* __restrict__ ei, int E,
                              const float* __restrict__ x,
                              float* __restrict__ agg) {
  int t = blockIdx.x * blockDim.x + threadIdx.x;
  int e = t >> 5;
  if (e >= E) return;
  int lane = t & 31;
  int src = ei[e];
  int dst = ei[E + e];
  if (dst >= N_G2) return;  // only first N_G2 rows feed the next layer
  float4 v = *(const float4*)&x[(size_t)src * 128 + lane * 4];
  float* a = &agg[(size_t)dst * 128 + lane * 4];
  atomicAdd(a + 0, v.x);
  atomicAdd(a + 1, v.y);
  atomicAdd(a + 2, v.z);
  atomicAdd(a + 3, v.w);
}

// h1 = relu(x + agg / max(deg,1)) , rows 0..N_G2
__global__ void k_concept_out(const float* __restrict__ x,
                              const float* __restrict__ agg,
                              const float* __restrict__ deg,
                              float* __restrict__ h) {
  int idx = blockIdx.x * blockDim.x + threadIdx.x;  // float4 index
  if (idx >= N_G2 * 32) return;
  int n = idx >> 5;
  float inv = 1.0f / fmaxf(deg[n], 1.0f);
  float4 xv = *(const float4*)&x[(size_t)idx * 4];
  float4 av = *(const float4*)&agg[(size_t)idx * 4];
  float4 r;
  r.x = fmaxf(fmaf(av.x, inv, xv.x), 0.0f);
  r.y = fmaxf(fmaf(av.y, inv, xv.y), 0.0f);
  r.z = fmaxf(fmaf(av.z, inv, xv.z), 0.0f);
  r.w = fmaxf(fmaf(av.w, inv, xv.w), 0.0f);
  *(float4*)&h[(size_t)idx * 4] = r;
}

// ----------------- pack B = [V reshaped (K x B*Od) | root] -----------------
// V shape (num_bases, K, Od); B column n<Nv -> V[n/Od][i][n%Od], else root.
__global__ void k_pack(const float* __restrict__ V, const float* __restrict__ root,
                       float* __restrict__ Bo, int K, int Ntot, int Nv, int Od) {
  int idx = blockIdx.x * blockDim.x + threadIdx.x;
  if (idx >= K * Ntot) return;
  int i = idx / Ntot, n = idx % Ntot;
  float v;
  if (n < Nv) {
    int b = n / Od, o = n % Od;
    v = V[((size_t)b * K + i) * Od + o];
  } else {
    v = root[(size_t)i * Od + (n - Nv)];
  }
  Bo[idx] = v;
}

// ----------------------------- f32 WMMA GEMM -------------------------------
// C(MxN) = A(MxK) * B(KxN); N % 32 == 0, K % 32 == 0.
// block = 256 threads = 8 waves; block tile 64x32; one 16x16 WMMA tile/wave.
// Interior blocks: K-tiles double-buffered via GLOBAL_LOAD_ASYNC_TO_LDS.
#define TM 64
#define TN 32
#define KC 32
#define SA 40
#define SB 40

__device__ __forceinline__ void g2l_b128(const float* g, float* l) {
  __builtin_amdgcn_global_load_async_to_lds_b128((AS1 v4i*)g, (AS3 v4i*)l, 0, 0);
}
__device__ __forceinline__ void g2l_b32(const float* g, float* l) {
  __builtin_amdgcn_global_load_async_to_lds_b32((AS1 int*)g, (AS3 int*)l, 0, 0);
}

__device__ __forceinline__ void wmma_tile(const float* Arow, const float* Brow,
                                          v8f& acc, int hi) {
#pragma unroll
  for (int k4 = 0; k4 < KC; k4 += 4) {
    int ko = k4 + hi * 2;
    v2f a, b;
    a.x = Arow[ko]; a.y = Arow[ko + 1];
    b.x = Brow[ko]; b.y = Brow[ko + 1];
    acc = __builtin_amdgcn_wmma_f32_16x16x4_f32(
        false, a, false, b, (short)0, acc, false, false);
  }
}

__global__ __launch_bounds__(256) void k_gemm_wmma(
    const float* __restrict__ A, const float* __restrict__ B,
    float* __restrict__ C, int M, int N, int K) {
  __shared__ __align__(16) float Al[2 * TM * SA];  // A tiles (dbl-buffered)
  __shared__ __align__(16) float Bl[2 * TN * SB];  // B tiles, transposed
  const int n0 = blockIdx.x * TN;
  const int m0 = blockIdx.y * TM;
  const int tid = threadIdx.x;
  const int lane = tid & 31;
  const int wave = tid >> 5;
  const int msub = wave >> 1;        // 0..3
  const int nsub = wave & 1;         // 0..1
  const int hi = lane >> 4;          // K-pair select (ISA f32 A/B layout)
  const int lo = lane & 15;          // M (A) / N (B) index within tile
  const int r  = tid >> 3;           // staging: row (A) / k (B), 0..31
  const int c4 = (tid & 7) * 4;      // staging: col group of 4

  v8f acc = {0.f, 0.f, 0.f, 0.f, 0.f, 0.f, 0.f, 0.f};

  if (m0 + TM <= M) {
    // ---------------- interior: async double-buffered staging --------------
    auto stage = [&](int kb, int buf) {
      float* AlB = &Al[buf * (TM * SA)];
      float* BlB = &Bl[buf * (TN * SB)];
      // A tile 64x32: 2 x b128 per thread, direct global->LDS
      g2l_b128(&A[(size_t)(m0 + r) * K + kb + c4], &AlB[r * SA + c4]);
      g2l_b128(&A[(size_t)(m0 + r + 32) * K + kb + c4], &AlB[(r + 32) * SA + c4]);
      // B tile 32x32 transposed: 4 x b32 per thread (per-lane LDS scatter)
      const float* gB = &B[(size_t)(kb + r) * N + n0 + c4];
      g2l_b32(gB + 0, &BlB[(c4 + 0) * SB + r]);
      g2l_b32(gB + 1, &BlB[(c4 + 1) * SB + r]);
      g2l_b32(gB + 2, &BlB[(c4 + 2) * SB + r]);
      g2l_b32(gB + 3, &BlB[(c4 + 3) * SB + r]);
    };
    stage(0, 0);
    const int nkb = K / KC;
    for (int i = 0; i < nkb; ++i) {
      const int cur = i & 1;
      if (i + 1 < nkb) {
        stage((i + 1) * KC, cur ^ 1);        // 6 async ops for next buffer
        __builtin_amdgcn_s_wait_asynccnt(6); // in-order: cur buffer complete
      } else {
        __builtin_amdgcn_s_wait_asynccnt(0);
      }
      __syncthreads();
      wmma_tile(&Al[cur * (TM * SA) + (msub * 16 + lo) * SA],
                &Bl[cur * (TN * SB) + (nsub * 16 + lo) * SB], acc, hi);
      __syncthreads();
    }
  } else {
    // ---------------- edge block: sync staging with M guard ----------------
    for (int kb = 0; kb < K; kb += KC) {
#pragma unroll
      for (int half = 0; half < 2; ++half) {
        int rr = r + half * 32;
        float4 v = make_float4(0.f, 0.f, 0.f, 0.f);
        int grow = m0 + rr;
        if (grow < M) v = *(const float4*)&A[(size_t)grow * K + kb + c4];
        float* d = &Al[rr * SA + c4];
        d[0] = v.x; d[1] = v.y; d[2] = v.z; d[3] = v.w;
      }
      float4 v = *(const float4*)&B[(size_t)(kb + r) * N + n0 + c4];
      Bl[(c4 + 0) * SB + r] = v.x;
      Bl[(c4 + 1) * SB + r] = v.y;
      Bl[(c4 + 2) * SB + r] = v.z;
      Bl[(c4 + 3) * SB + r] = v.w;
      __syncthreads();
      wmma_tile(&Al[(msub * 16 + lo) * SA], &Bl[(nsub * 16 + lo) * SB], acc, hi);
      __syncthreads();
    }
  }

  const int col = n0 + nsub * 16 + lo;
#pragma unroll
  for (int v = 0; v < 8; ++v) {
    int row = m0 + msub * 16 + hi * 8 + v;
    if (row < M) C[(size_t)row * N + col] = acc[v];
  }
}

// ------------------------- edge messages, layer 1 --------------------------
// one wave per edge; 64 outputs -> 2 per lane (float2 gather, coalesced)
__global__ void k_edge_msg1(const int* __restrict__ ei, const int* __restrict__ et,
                            int E, const float* __restrict__ comp,
                            const float* __restrict__ P,  // ldp = 576
                            float* __restrict__ agg) {
  int t = blockIdx.x * blockDim.x + threadIdx.x;
  int e = t >> 5;
  if (e >= E) return;
  int lane = t & 31;
  int src = ei[e];
  int dst = ei[E + e];
  int r = et[e];
  const float* prow = &P[(size_t)src * 576];
  const float* c = &comp[(size_t)r * 8];
  float a0 = 0.f, a1 = 0.f;
#pragma unroll
  for (int b = 0; b < 8; ++b) {
    float cb = c[b];
    float2 p = *(const float2*)&prow[b * 64 + lane * 2];
    a0 = fmaf(cb, p.x, a0);
    a1 = fmaf(cb, p.y, a1);
  }
  float* a = &agg[(size_t)dst * 64 + lane * 2];
  atomicAdd(a + 0, a0);
  atomicAdd(a + 1, a1);
}

// h2 = relu(agg/deg + P[:,512:576] + bias1)
__global__ void k_layer1_out(const float* __restrict__ agg,
                             const float* __restrict__ deg,
                             const float* __restrict__ P,  // ldp = 576
                             const float* __restrict__ bias,
                             float* __restrict__ h) {
  int idx = blockIdx.x * blockDim.x + threadIdx.x;
  if (idx >= N_G2 * 64) return;
  int n = idx >> 6;
  int o = idx & 63;
  float inv = 1.0f / fmaxf(deg[n], 1.0f);
  float v = fmaf(agg[idx], inv, P[(size_t)n * 576 + 512 + o] + bias[o]);
  h[idx] = fmaxf(v, 0.0f);
}

// ------------------------- edge messages, layer 2 --------------------------
// one wave per edge; 32 outputs -> 1 per lane
__global__ void k_edge_msg2(const int* __restrict__ ei, const int* __restrict__ et,
                            int E, const float* __restrict__ comp,
                            const float* __restrict__ P,  // ldp = 288
                            float* __restrict__ agg) {
  int t = blockIdx.x * blockDim.x + threadIdx.x;
  int e = t >> 5;
  if (e >= E) return;
  int lane = t & 31;
  int src = ei[e];
  int dst = ei[E + e];
  int r = et[e];
  const float* prow = &P[(size_t)src * 288];
  const float* c = &comp[(size_t)r * 8];
  float a0 = 0.f;
#pragma unroll
  for (int b = 0; b < 8; ++b)
    a0 = fmaf(c[b], prow[b * 32 + lane], a0);
  atomicAdd(&agg[(size_t)dst * 32 + lane], a0);
}

// final: z = agg/deg + P[:,256:288] + bias2 ; softmax over 32 classes
__global__ void k_final(const float* __restrict__ agg,
                        const float* __restrict__ deg,
                        const float* __restrict__ P,  // ldp = 288
                        const float* __restrict__ bias,
                        float* __restrict__ out) {
  int t = blockIdx.x * blockDim.x + threadIdx.x;
  int row = t >> 5;
  if (row >= N_G2) return;
  int lane = t & 31;
  float inv = 1.0f / fmaxf(deg[row], 1.0f);
  float z = fmaf(agg[(size_t)row * 32 + lane], inv,
                 P[(size_t)row * 288 + 256 + lane] + bias[lane]);
  float m = z;
#pragma unroll
  for (int off = 16; off > 0; off >>= 1) m = fmaxf(m, __shfl_xor(m, off, 32));
  float ez = __expf(z - m);
  float s = ez;
#pragma unroll
  for (int off = 16; off > 0; off >>= 1) s += __shfl_xor(s, off, 32);
  out[(size_t)row * 32 + lane] = ez / s;
}

// ---------------------------------------------------------------------------
extern "C" void kernel_launch(void* const* d_in, const int* in_sizes, int n_in,
                              void* d_out, int out_size, void* d_ws, size_t ws_size,
                              hipStream_t stream) {
  (void)n_in; (void)out_size; (void)ws_size;
  const int*   ei_g2 = (const int*)d_in[0];
  const int*   et_g2 = (const int*)d_in[1];
  const int*   ei_g1 = (const int*)d_in[2];
  const float* x0    = (const float*)d_in[3];
  const float* V1    = (const float*)d_in[4];
  const float* comp1 = (const float*)d_in[5];
  const float* root1 = (const float*)d_in[6];
  const float* bias1 = (const float*)d_in[7];
  const float* V2    = (const float*)d_in[8];
  const float* comp2 = (const float*)d_in[9];
  const float* root2 = (const float*)d_in[10];
  const float* bias2 = (const float*)d_in[11];
  float* out = (float*)d_out;
  const int E2 = in_sizes[0] / 2;
  const int E1 = in_sizes[2] / 2;

  float* ws = (float*)d_ws;
  // workspace layout (floats), with aliasing:
  const size_t o_deg1 = 0;                                   // 100000
  const size_t o_aggc = (size_t)N_G2;                        // 12.8 M
  const size_t o_agg1 = o_aggc;                              // (reuse) 6.4 M
  const size_t o_agg2 = o_aggc + (size_t)N_G2 * 64;          // (reuse) 3.2 M
  const size_t o_deg2 = o_agg2 + (size_t)N_G2 * 32;          // (reuse) 100000
  const size_t o_h1   = o_aggc + (size_t)N_G2 * 128;         // 12.8 M
  const size_t o_h2   = o_h1;                                // (reuse) 6.4 M
  const size_t o_B1   = o_h1 + (size_t)N_G2 * 128;           // 128*576
  const size_t o_B2   = o_B1 + 128 * 576;                    // 64*288
  const size_t o_P1   = o_B2 + 64 * 288;                     // 100032*576
  const size_t o_P2   = o_P1;                                // (reuse) 100032*288

  // ---- concept layer on G1 ----
  (void)hipMemsetAsync(ws + o_deg1, 0, (o_h1 - o_deg1) * sizeof(float), stream);
  k_degree<<<(E1 + 255) / 256, 256, 0, stream>>>(ei_g1, E1, ws + o_deg1, N_G2);
  k_concept_agg<<<(E1 * 32 + 255) / 256, 256, 0, stream>>>(ei_g1, E1, x0, ws + o_aggc);
  k_concept_out<<<(N_G2 * 32 + 255) / 256, 256, 0, stream>>>(x0, ws + o_aggc,
                                                             ws + o_deg1, ws + o_h1);

  // ---- RGCN layer 1 ----
  (void)hipMemsetAsync(ws + o_agg1, 0,
                       ((size_t)N_G2 * 96 + (size_t)N_G2) * sizeof(float), stream);
  k_degree<<<(E2 + 255) / 256, 256, 0, stream>>>(ei_g2, E2, ws + o_deg2, N_G2);
  k_pack<<<(128 * 576 + 255) / 256, 256, 0, stream>>>(V1, root1, ws + o_B1,
                                                      128, 576, 512, 64);
  dim3 g1(576 / TN, (N_G2 + TM - 1) / TM);
  k_gemm_wmma<<<g1, 256, 0, stream>>>(ws + o_h1, ws + o_B1, ws + o_P1,
                                      N_G2, 576, 128);
  k_edge_msg1<<<(E2 * 32 + 255) / 256, 256, 0, stream>>>(ei_g2, et_g2, E2, comp1,
                                                         ws + o_P1, ws + o_agg1);
  k_layer1_out<<<(N_G2 * 64 + 255) / 256, 256, 0, stream>>>(ws + o_agg1, ws + o_deg2,
                                                            ws + o_P1, bias1, ws + o_h2);

  // ---- RGCN layer 2 + softmax ----
  k_pack<<<(64 * 288 + 255) / 256, 256, 0, stream>>>(V2, root2, ws + o_B2,
                                                     64, 288, 256, 32);
  dim3 g2(288 / TN, (N_G2 + TM - 1) / TM);
  k_gemm_wmma<<<g2, 256, 0, stream>>>(ws + o_h2, ws + o_B2, ws + o_P2,
                                      N_G2, 288, 64);
  k_edge_msg2<<<(E2 * 32 + 255) / 256, 256, 0, stream>>>(ei_g2, et_g2, E2, comp2,
                                                         ws + o_P2, ws + o_agg2);
  k_final<<<(N_G2 * 32 + 255) / 256, 256, 0, stream>>>(ws + o_agg2, ws + o_deg2,
                                                       ws + o_P2, bias2, out);
}